// DSNTJSDDistanceDoubleLossNew_20083267076467
// MI455X (gfx1250) — compile-verified
//
#include <hip/hip_runtime.h>
#include <stdint.h>

// Problem constants (B=128, C=2, H=W=256)
namespace {
constexpr int kW        = 256;
constexpr int kChunk    = 4096;                 // elements per block
constexpr int kBlkPerHm = 16;                   // 65536 / 4096
constexpr int kNHm      = 256;                  // B*C heatmaps
constexpr int kThreads  = 256;
constexpr int kGrid     = kNHm * kBlkPerHm;     // 4096 blocks
}

typedef unsigned int u32x4 __attribute__((ext_vector_type(4)));
typedef int          i32x8 __attribute__((ext_vector_type(8)));
typedef int          i32x4 __attribute__((ext_vector_type(4)));

// ---------------------------------------------------------------------------
// Pass 1: per-chunk online-softmax partials of `input` (running max, sum-exp)
// and argmax partials of `target`. Plain coalesced float4 streaming; this pass
// is the HBM-bound pass and also populates the 192MB L2 for pass 2.
// ---------------------------------------------------------------------------
__global__ __launch_bounds__(kThreads)
void dsnt_pass1(const float* __restrict__ in, const float* __restrict__ tg,
                float4* __restrict__ part1) {
  const int blk = blockIdx.x;          // 0..4095
  const int t   = threadIdx.x;
  const int chunkInHm = blk & (kBlkPerHm - 1);
  const size_t base = (size_t)blk * kChunk;

  const float4* in4 = (const float4*)(in + base);
  const float4* tg4 = (const float4*)(tg + base);

  // 16 input elements per thread: 4 x float4, stride-256 float4s (coalesced)
  float xs[16];
#pragma unroll
  for (int k = 0; k < 4; ++k) {
    float4 v = in4[k * kThreads + t];
    xs[4*k+0] = v.x; xs[4*k+1] = v.y; xs[4*k+2] = v.z; xs[4*k+3] = v.w;
  }
  float xm = -3.0e38f;
#pragma unroll
  for (int i = 0; i < 16; ++i) xm = fmaxf(xm, xs[i]);
  float se = 0.0f;
#pragma unroll
  for (int i = 0; i < 16; ++i) se += __expf(xs[i] - xm);

  // target argmax over the same 16 elements (first-occurrence semantics:
  // indices are visited in increasing order, update on strictly-greater)
  float tv = -3.0e38f;
  int   ti = 0;
#pragma unroll
  for (int k = 0; k < 4; ++k) {
    float4 v = tg4[k * kThreads + t];
    const int e0 = (k * kThreads + t) * 4;
    if (v.x > tv) { tv = v.x; ti = e0 + 0; }
    if (v.y > tv) { tv = v.y; ti = e0 + 1; }
    if (v.z > tv) { tv = v.z; ti = e0 + 2; }
    if (v.w > tv) { tv = v.w; ti = e0 + 3; }
  }
  int tiHm = chunkInHm * kChunk + ti;  // index within the heatmap

  __shared__ float sm[kThreads], ss[kThreads], stv[kThreads];
  __shared__ int   sti[kThreads];
  sm[t] = xm; ss[t] = se; stv[t] = tv; sti[t] = tiHm;
  __syncthreads();
  for (int off = kThreads / 2; off > 0; off >>= 1) {
    if (t < off) {
      float m2 = sm[t + off], s2 = ss[t + off];
      float M  = fmaxf(sm[t], m2);
      ss[t] = ss[t] * __expf(sm[t] - M) + s2 * __expf(m2 - M);
      sm[t] = M;
      float v2 = stv[t + off]; int i2 = sti[t + off];
      if (v2 > stv[t] || (v2 == stv[t] && i2 < sti[t])) { stv[t] = v2; sti[t] = i2; }
    }
    __syncthreads();
  }
  if (t == 0)
    part1[blk] = make_float4(sm[0], ss[0], stv[0], __int_as_float(sti[0]));
}

// ---------------------------------------------------------------------------
// Combine pass-1 partials: per heatmap -> {max M, 1/Z, true_x, true_y}
// ---------------------------------------------------------------------------
__global__ __launch_bounds__(kNHm)
void dsnt_combine1(const float4* __restrict__ part1, float4* __restrict__ hm) {
  const int h = threadIdx.x;           // heatmap id 0..255
  float M = -3.0e38f, S = 0.0f, tv = -3.0e38f;
  int   ti = 0;
  for (int i = 0; i < kBlkPerHm; ++i) {
    float4 p = part1[h * kBlkPerHm + i];
    float Mn = fmaxf(M, p.x);
    S = S * __expf(M - Mn) + p.y * __expf(p.x - Mn);
    M = Mn;
    int i2 = __float_as_int(p.w);
    if (p.z > tv || (p.z == tv && i2 < ti)) { tv = p.z; ti = i2; }
  }
  const float invZ = 1.0f / S;
  const float tx = ((float)(ti & (kW - 1)) - 127.0f) * (1.0f / 256.0f);
  const float ty = ((float)(ti >> 8)       - 127.0f) * (1.0f / 256.0f);
  hm[h] = make_float4(M, invZ, tx, ty);
}

// ---------------------------------------------------------------------------
// Pass 2: stream both tensors again (L2-resident) via the Tensor Data Mover.
// Wave 0 issues two TDM descriptors (input chunk, target chunk -> LDS),
// waits on TENSORcnt, then the workgroup consumes the tiles from LDS:
//   p   = exp(x - M) / Z
//   acc_px += p*(col-127); acc_py += p*(row-127)      (scaled by 1/256 later)
//   m = (t+p)/2 ; acc_jsd += m*log(m) - m*m           (fused 0.5*kl+0.5*kl)
// ---------------------------------------------------------------------------
__global__ __launch_bounds__(kThreads)
void dsnt_pass2(const float* __restrict__ in, const float* __restrict__ tg,
                const float4* __restrict__ hm, float4* __restrict__ part2) {
  __shared__ float lin[kChunk];
  __shared__ float ltg[kChunk];
  __shared__ float r0[kThreads], r1[kThreads], r2[kThreads];

  const int blk = blockIdx.x;
  const int t   = threadIdx.x;
  const int h   = blk >> 4;            // heatmap
  const int chunkInHm = blk & (kBlkPerHm - 1);

  const float4 P   = hm[h];
  const float  M   = P.x;
  const float  invZ = P.y;

  // --- TDM stage: one wave issues the async tensor loads ------------------
  if (t < 32) {
    const uint64_t gin = (uint64_t)(uintptr_t)(in + (size_t)blk * kChunk);
    const uint64_t gtg = (uint64_t)(uintptr_t)(tg + (size_t)blk * kChunk);
    const uint32_t loff_in = (uint32_t)(uintptr_t)&lin[0];  // LDS byte offset
    const uint32_t loff_tg = (uint32_t)(uintptr_t)&ltg[0];

    // D# group0: {count=1} | lds_addr | global_addr[31:0] | ga[56:32]+type=2
    u32x4 g0i, g0t;
    g0i[0] = 1u;
    g0i[1] = loff_in;
    g0i[2] = (uint32_t)gin;
    g0i[3] = ((uint32_t)(gin >> 32) & 0x1FFFFFFu) | 0x80000000u;
    g0t[0] = 1u;
    g0t[1] = loff_tg;
    g0t[2] = (uint32_t)gtg;
    g0t[3] = ((uint32_t)(gtg >> 32) & 0x1FFFFFFu) | 0x80000000u;

    // D# group1: data_size=4B, tensor_dim0=4096, tensor_dim1=1,
    //            tile_dim0=4096, tile_dim1=1, dim0_stride=4096,
    //            workgroup_mask=0 (not in a cluster), no pad/iterate/barrier
    i32x8 g1;
    g1[0] = (int)(2u << 16);            // data_size = 4 bytes
    g1[1] = (int)(4096u << 16);         // tensor_dim0[15:0]
    g1[2] = (int)(1u << 16);            // tensor_dim1 = 1
    g1[3] = (int)(4096u << 16);         // tile_dim0 = 4096
    g1[4] = 1;                          // tile_dim1 = 1
    g1[5] = 4096;                       // tensor_dim0_stride
    g1[6] = 0;
    g1[7] = 0;

    i32x4 z4 = {0, 0, 0, 0};            // groups 2/3 unused (<=2D tensor)
    i32x8 z8 = {0, 0, 0, 0, 0, 0, 0, 0};// extra descriptor words: unused

    __builtin_amdgcn_tensor_load_to_lds(g0i, g1, z4, z4, z8, 0);
    __builtin_amdgcn_tensor_load_to_lds(g0t, g1, z4, z4, z8, 0);
    __builtin_amdgcn_s_wait_tensorcnt(0);
  }
  __syncthreads();

  // --- consume the staged tiles from LDS -----------------------------------
  const int gbase = chunkInHm * kChunk;  // element offset within heatmap
  const float4* lin4 = (const float4*)lin;
  const float4* ltg4 = (const float4*)ltg;

  float apx = 0.0f, apy = 0.0f, aj = 0.0f;
#pragma unroll
  for (int k = 0; k < 4; ++k) {
    const int li = k * kThreads + t;     // float4 index in chunk
    float4 xv = lin4[li];
    float4 tv = ltg4[li];
    const int e0 = gbase + li * 4;
    float xe[4] = {xv.x, xv.y, xv.z, xv.w};
    float te[4] = {tv.x, tv.y, tv.z, tv.w};
#pragma unroll
    for (int j = 0; j < 4; ++j) {
      const int g = e0 + j;
      const float p   = __expf(xe[j] - M) * invZ;
      const float col = (float)(g & (kW - 1));
      const float row = (float)(g >> 8);
      apx += p * (col - 127.0f);
      apy += p * (row - 127.0f);
      const float mm = 0.5f * (te[j] + p);
      aj += mm * __logf(mm) - mm * mm;   // == 0.5*kl(gt,m)+0.5*kl(pred,m) elt
    }
  }

  r0[t] = apx; r1[t] = apy; r2[t] = aj;
  __syncthreads();
  for (int off = kThreads / 2; off > 0; off >>= 1) {
    if (t < off) { r0[t] += r0[t+off]; r1[t] += r1[t+off]; r2[t] += r2[t+off]; }
    __syncthreads();
  }
  if (t == 0)
    part2[blk] = make_float4(r0[0] * (1.0f / 256.0f),
                             r1[0] * (1.0f / 256.0f), r2[0], 0.0f);
}

// ---------------------------------------------------------------------------
// Final: per-heatmap ed + jsd, inter-landmark term per batch, scalar s/B.
// ---------------------------------------------------------------------------
__global__ __launch_bounds__(kNHm)
void dsnt_final(const float4* __restrict__ part2, const float4* __restrict__ hm,
                float* __restrict__ out) {
  __shared__ float spx[kNHm], spy[kNHm], stx[kNHm], sty[kNHm], sterm[kNHm];
  const int h = threadIdx.x;

  float px = 0.0f, py = 0.0f, js = 0.0f;
  for (int i = 0; i < kBlkPerHm; ++i) {
    float4 p = part2[h * kBlkPerHm + i];
    px += p.x; py += p.y; js += p.z;
  }
  const float4 P = hm[h];
  const float tx = P.z, ty = P.w;
  const float jsd = js * (1.0f / 65536.0f);       // mean over H*W
  const float dx = tx - px, dy = ty - py;
  float term = sqrtf(dx * dx + dy * dy) + jsd;    // ed + jsd

  spx[h] = px; spy[h] = py; stx[h] = tx; sty[h] = ty;
  __syncthreads();

  if (h < 128) {                                   // one per batch element
    const int c0 = 2 * h, c1 = 2 * h + 1;
    const float pdx = spx[c0] - spx[c1], pdy = spy[c0] - spy[c1];
    const float tdx = stx[c0] - stx[c1], tdy = sty[c0] - sty[c1];
    const float pd = sqrtf(pdx * pdx + pdy * pdy);
    const float td = sqrtf(tdx * tdx + tdy * tdy);
    term += fabsf(pd - td) / td;                   // sqrt((pd-td)^2)/td
  }
  sterm[h] = term;
  __syncthreads();
  for (int off = kNHm / 2; off > 0; off >>= 1) {
    if (h < off) sterm[h] += sterm[h + off];
    __syncthreads();
  }
  if (h == 0) out[0] = sterm[0] * (1.0f / 128.0f); // s / B
}

// ---------------------------------------------------------------------------
extern "C" void kernel_launch(void* const* d_in, const int* in_sizes, int n_in,
                              void* d_out, int out_size, void* d_ws, size_t ws_size,
                              hipStream_t stream) {
  const float* in = (const float*)d_in[0];
  const float* tg = (const float*)d_in[1];
  float* out = (float*)d_out;

  // workspace layout (float4): part1[4096] | hm[256] | part2[4096]  (~133 KiB)
  float4* part1 = (float4*)d_ws;
  float4* hm    = part1 + kGrid;
  float4* part2 = hm + kNHm;

  dsnt_pass1  <<<kGrid, kThreads, 0, stream>>>(in, tg, part1);
  dsnt_combine1<<<1,    kNHm,     0, stream>>>(part1, hm);
  dsnt_pass2  <<<kGrid, kThreads, 0, stream>>>(in, tg, hm, part2);
  dsnt_final  <<<1,     kNHm,     0, stream>>>(part2, hm, out);
}